// UnPool2D_86217173500128
// MI455X (gfx1250) — compile-verified
//
#include <hip/hip_runtime.h>
#include <cstdint>

// Zero-stuffed 2x depthwise upsample: out[b,c,2i,2j] = x[b,c,i,j], else 0.
// Input  (32,64,128,128) f32  = 128 MiB read
// Output (32,64,256,256) f32  = 512 MiB write   -> HBM-bound, ~29 us @ 23.3 TB/s
//
// Even output rows: build {x0,0,x1,0} in VGPRs, store b128 non-temporal.
// Odd  output rows: all-zero -> CDNA5 async DMA store from a zeroed LDS slot
//                   (global_store_async_from_lds_b128, ASYNCcnt), bypassing VGPR data.

typedef float __attribute__((ext_vector_type(4))) v4f;
typedef float __attribute__((ext_vector_type(2))) v2f;

__global__ __launch_bounds__(256) void upsample2x_zero_fill(
    const float* __restrict__ x, float* __restrict__ out) {
  __shared__ __align__(16) float zbuf[256 * 4];

  const int tid = threadIdx.x;

  // Stage a private 16B zero slot in LDS for this lane (ds_store_b128).
  v4f z4 = {0.f, 0.f, 0.f, 0.f};
  *reinterpret_cast<v4f*>(&zbuf[tid * 4]) = z4;
  // Flat LDS address bits [31:0] are the LDS byte offset (ISA 10.2 aperture rules).
  const uint32_t lds_addr = (uint32_t)(uintptr_t)(&zbuf[tid * 4]);

  // One output float4 per thread.
  const int idx4   = blockIdx.x * 256 + tid;  // float4 index into output
  const int col4   = idx4 & 63;               // 256 floats per row -> 64 float4
  const int rowAll = idx4 >> 6;               // row index across (b*c*Hout)
  const int row    = rowAll & 255;            // row within image (Hout = 256)
  const int img    = rowAll >> 8;             // flattened b*c image index

  float* op = out + ((long long)idx4 << 2);

  if (row & 1) {
    // Odd output row: pure zeros. Ensure this lane's LDS zero-slot store has
    // landed, then DMA LDS -> global (no VGPR data movement).
    asm volatile("s_wait_dscnt 0" ::: "memory");
    asm volatile("global_store_async_from_lds_b128 %0, %1, off"
                 :: "v"((void*)op), "v"(lds_addr)
                 : "memory");
  } else {
    // Even output row: cols [4c..4c+3] <- {x[2c], 0, x[2c+1], 0}.
    const v2f* __restrict__ xin = reinterpret_cast<const v2f*>(x);
    const v2f v = __builtin_nontemporal_load(
        &xin[(img * 128 + (row >> 1)) * 64 + col4]);  // b64, NT (read-once)
    v4f o = {v.x, 0.f, v.y, 0.f};
    __builtin_nontemporal_store(o, reinterpret_cast<v4f*>(op));  // b128, NT
  }
}

extern "C" void kernel_launch(void* const* d_in, const int* in_sizes, int n_in,
                              void* d_out, int out_size, void* d_ws, size_t ws_size,
                              hipStream_t stream) {
  (void)in_sizes; (void)n_in; (void)d_ws; (void)ws_size;
  const float* x = (const float*)d_in[0];
  float* out = (float*)d_out;

  const int n4 = out_size / 4;        // 33,554,432 float4 stores
  dim3 block(256);
  dim3 grid(n4 / 256);                // 131,072 blocks (out_size is 2^27)
  hipLaunchKernelGGL(upsample2x_zero_fill, grid, block, 0, stream, x, out);
}